// AdversarialHead_32212254720153
// MI455X (gfx1250) — compile-verified
//
#include <hip/hip_runtime.h>
#include <hip/hip_bf16.h>

typedef __attribute__((ext_vector_type(2))) float v2f;
typedef __attribute__((ext_vector_type(8))) float v8f;
typedef __attribute__((ext_vector_type(4))) int   v4i;

#define FEAT   128
#define IN1    149          // 128 + 21
#define IN1P   152          // padded K for fc1
#define HIDN   140
#define HIDP   144          // padded (N for fc1, K for fc2)
#define INV    256          // 2*FEAT, K for heads
#define AOHW   24           // padded one-hot width (21 -> 24)
#define NROWS  (4096 * 32)  // 131072 flattened rows
#define BLKS   4            // 16-row blocks per workgroup (2 waves each)
#define TPB    (BLKS * 64)  // 256 threads = 8 waves -> 2 waves/SIMD32

// ---- LDS layout (units: floats), allocated dynamically ----
#define W1_OFF   0
#define W1_SZ    (144 * IN1P)           // fc1_w padded [144][152]
#define W2_OFF   (W1_OFF + W1_SZ)
#define W2_SZ    (128 * HIDP)           // fc2_w padded [128][144]
#define WH_OFF   (W2_OFF + W2_SZ)
#define WH_SZ    (32 * INV)             // stacked heads padded [32][256]
#define CUR_OFF  (WH_OFF + WH_SZ)       // per-block [16][128]
#define NXT_OFF  (CUR_OFF + BLKS * 16 * FEAT)
#define AOH_OFF  (NXT_OFF + BLKS * 16 * FEAT)   // per-block [16][24]
#define H_OFF    (AOH_OFF + BLKS * 16 * AOHW)   // per-block [16][144]
#define LDS_FLOATS (H_OFF + BLKS * 16 * HIDP)   // 75,648 floats = 302,592 B

#if defined(__gfx1250__) && \
    __has_builtin(__builtin_amdgcn_global_load_async_to_lds_b128) && \
    __has_builtin(__builtin_amdgcn_s_wait_asynccnt)
#define USE_ASYNC 1
#else
#define USE_ASYNC 0
#endif

__device__ __forceinline__ v8f wmma4(v2f a, v2f b, v8f c) {
  // D = A(16x4 f32) * B(4x16 f32) + C(16x16 f32)
  return __builtin_amdgcn_wmma_f32_16x16x4_f32(
      /*neg_a=*/false, a, /*neg_b=*/false, b,
      /*c_mod=*/(short)0, c, /*reuse_a=*/false, /*reuse_b=*/false);
}

#if USE_ASYNC
typedef __attribute__((address_space(1))) v4i* gv4i_p;   // global (AS1)
typedef __attribute__((address_space(3))) v4i* lv4i_p;   // LDS (AS3)
__device__ __forceinline__ void async_cp16(const float4* g, float4* l) {
  __builtin_amdgcn_global_load_async_to_lds_b128(
      (gv4i_p)(void*)g, (lv4i_p)(void*)l, /*offset=*/0, /*cpol=*/0);
}
#endif

__device__ __forceinline__ void fc1_store(float* HB, const float* fc1_b,
                                          int n0, int arow, int hi, v8f acc) {
  int n = n0 + arow;
  float bias = (n < HIDN) ? fc1_b[n] : 0.f;
#pragma unroll
  for (int j = 0; j < 8; ++j) {          // C VGPR j: M = j + 8*hi, N = arow
    float v = acc[j] + bias;
    v = (v > 0.f) ? v : 0.1f * v;        // LeakyReLU(0.1)
    HB[(j + 8 * hi) * HIDP + n] = v;
  }
}

__device__ __forceinline__ void fc2_store(float* out, const float* fc2_b,
                                          int r0, int n0, int arow, int hi, v8f acc) {
  int n = n0 + arow;
  float bias = fc2_b[n];
#pragma unroll
  for (int j = 0; j < 8; ++j)
    out[(size_t)(r0 + j + 8 * hi) * FEAT + n] = acc[j] + bias;
}

extern "C" __global__ __launch_bounds__(TPB, 1)
void icm_fused_kernel(const float* __restrict__ cur,
                      const float* __restrict__ nxt,
                      const int*   __restrict__ actions,
                      const float* __restrict__ fc1_w, const float* __restrict__ fc1_b,
                      const float* __restrict__ fc2_w, const float* __restrict__ fc2_b,
                      const float* __restrict__ hw0, const float* __restrict__ hb0,
                      const float* __restrict__ hw1, const float* __restrict__ hb1,
                      const float* __restrict__ hw2, const float* __restrict__ hb2,
                      float* __restrict__ out)
{
  extern __shared__ float lds[];
  const int tid   = threadIdx.x;
  const int lane  = tid & 31;
  const int half  = (tid >> 5) & 1;                       // wave within pair
  const int uhalf = __builtin_amdgcn_readfirstlane(half); // scalar branch key
  const int blkL  = tid >> 6;                             // row-block 0..3

  // ---------- stage zero-padded weights into LDS (whole workgroup) ----------
  for (int i = tid; i < W1_SZ; i += TPB) {
    int n = i / IN1P, k = i - n * IN1P;
    lds[W1_OFF + i] = (n < HIDN && k < IN1) ? fc1_w[n * IN1 + k] : 0.f;
  }
  for (int i = tid; i < W2_SZ; i += TPB) {
    int n = i / HIDP, k = i - n * HIDP;
    lds[W2_OFF + i] = (k < HIDN) ? fc2_w[n * HIDN + k] : 0.f;
  }
  for (int i = tid; i < WH_SZ; i += TPB) {
    int n = i >> 8, k = i & 255;
    float v = 0.f;
    if      (n < 5)  v = hw0[n * INV + k];
    else if (n < 12) v = hw1[(n - 5) * INV + k];
    else if (n < 21) v = hw2[(n - 12) * INV + k];
    lds[WH_OFF + i] = v;
  }

  // ---------- per-block activation staging (2 waves share one block) -------
  const int blk = blockIdx.x * BLKS + blkL;   // 0 .. 8191
  const int r0  = blk * 16;

  float* CUR = lds + CUR_OFF + blkL * (16 * FEAT);
  float* NXT = lds + NXT_OFF + blkL * (16 * FEAT);
  float* AOH = lds + AOH_OFF + blkL * (16 * AOHW);
  float* HB  = lds + H_OFF   + blkL * (16 * HIDP);

  {
    const float4* gc = (const float4*)(cur + (size_t)r0 * FEAT);
    const float4* gn = (const float4*)(nxt + (size_t)r0 * FEAT);
    float4* lc = (float4*)CUR;
    float4* ln = (float4*)NXT;
#pragma unroll
    for (int i = 0; i < 8; ++i) {        // each wave moves half of each tile
      int idx = (half * 8 + i) * 32 + lane;
#if USE_ASYNC
      async_cp16(gc + idx, lc + idx);
      async_cp16(gn + idx, ln + idx);
#else
      lc[idx] = gc[idx];
      ln[idx] = gn[idx];
#endif
    }
  }

  // multi-hot action encoding (wave 0 of pair only: in-wave LDS ordering)
  if (uhalf == 0) {
#pragma unroll
    for (int i = 0; i < (16 * AOHW) / 32; ++i) AOH[i * 32 + lane] = 0.f;
    if (lane < 16) {
      int r = r0 + lane;
      int f = r >> 5, a = r & 31;
      const int* ap = actions + (size_t)f * 96 + a;   // [F,3,A]
      AOH[lane * AOHW + ap[0]]       = 1.f;           // type0: +0
      AOH[lane * AOHW + ap[32] + 5]  = 1.f;           // type1: +5
      AOH[lane * AOHW + ap[64] + 12] = 1.f;           // type2: +12
    }
  }

#if USE_ASYNC
  __builtin_amdgcn_s_wait_asynccnt(0);
#endif
  __syncthreads();   // weights + activations visible to both waves of pair

  const int arow = lane & 15;     // A row / B column (relative)
  const int hi   = lane >> 4;     // lanes 16-31 hold K+2,K+3 of each K-quad
  const int kofs = 2 * hi;

  // ---------- fc1: [16 x 152] @ [152 x 144] -> LeakyReLU -> HB ------------
  // wave-half 0: n-tiles {0,1},{2,3},{8};  wave-half 1: {4,5},{6,7}
#pragma unroll
  for (int np = 0; np < 2; ++np) {
    const int n0 = (uhalf * 4 + np * 2) * 16;
    const float* wb0 = lds + W1_OFF + (n0 + arow) * IN1P + kofs;
    const float* wb1 = wb0 + 16 * IN1P;
    v8f acc0 = {}, acc1 = {};
    for (int kb = 0; kb < FEAT; kb += 4) {
      v2f a = *(const v2f*)(CUR + arow * FEAT + kb + kofs);
      acc0 = wmma4(a, *(const v2f*)(wb0 + kb), acc0);
      acc1 = wmma4(a, *(const v2f*)(wb1 + kb), acc1);
    }
    for (int kb = FEAT; kb < IN1P; kb += 4) {
      v2f a = *(const v2f*)(AOH + arow * AOHW + (kb - FEAT) + kofs);
      acc0 = wmma4(a, *(const v2f*)(wb0 + kb), acc0);
      acc1 = wmma4(a, *(const v2f*)(wb1 + kb), acc1);
    }
    fc1_store(HB, fc1_b, n0,      arow, hi, acc0);
    fc1_store(HB, fc1_b, n0 + 16, arow, hi, acc1);
  }
  if (uhalf == 0) {                      // tail tile nt=8 (n = 128..143)
    const int n0 = 128;
    const float* wb = lds + W1_OFF + (n0 + arow) * IN1P + kofs;
    v8f acc = {};
    for (int kb = 0; kb < FEAT; kb += 4)
      acc = wmma4(*(const v2f*)(CUR + arow * FEAT + kb + kofs),
                  *(const v2f*)(wb + kb), acc);
    for (int kb = FEAT; kb < IN1P; kb += 4)
      acc = wmma4(*(const v2f*)(AOH + arow * AOHW + (kb - FEAT) + kofs),
                  *(const v2f*)(wb + kb), acc);
    fc1_store(HB, fc1_b, n0, arow, hi, acc);
  }

  // ---------- heads (independent of HB; runs before the barrier) ----------
  const size_t L0 = (size_t)NROWS * FEAT;
  const size_t L1 = L0 + (size_t)NROWS * 5;
  const size_t L2 = L1 + (size_t)NROWS * 7;
  {
    const int n0 = uhalf * 16;           // one head tile per wave of pair
    const float* wb = lds + WH_OFF + (n0 + arow) * INV + kofs;
    v8f acc = {};
    for (int kb = 0; kb < FEAT; kb += 4)
      acc = wmma4(*(const v2f*)(CUR + arow * FEAT + kb + kofs),
                  *(const v2f*)(wb + kb), acc);
    for (int kb = FEAT; kb < INV; kb += 4)
      acc = wmma4(*(const v2f*)(NXT + arow * FEAT + (kb - FEAT) + kofs),
                  *(const v2f*)(wb + kb), acc);
    int c = n0 + arow;                   // global head column (valid < 21)
    float bias = 0.f; size_t base = 0; int width = 0, coff = 0; bool valid = true;
    if      (c < 5)  { bias = hb0[c];      base = L0; width = 5; coff = c;      }
    else if (c < 12) { bias = hb1[c - 5];  base = L1; width = 7; coff = c - 5;  }
    else if (c < 21) { bias = hb2[c - 12]; base = L2; width = 9; coff = c - 12; }
    else valid = false;
    if (valid) {
#pragma unroll
      for (int j = 0; j < 8; ++j) {
        int r = r0 + j + 8 * hi;
        out[base + (size_t)r * width + coff] = acc[j] + bias;
      }
    }
  }

  __syncthreads();   // HB (written by both waves of pair) ready for fc2

  // ---------- fc2: [16 x 144] @ [144 x 128] -> next_feature_pred ----------
  // wave-half 0: n-tiles {0,1},{2,3};  wave-half 1: {4,5},{6,7}
#pragma unroll
  for (int np = 0; np < 2; ++np) {
    const int n0 = (uhalf * 4 + np * 2) * 16;
    const float* wb0 = lds + W2_OFF + (n0 + arow) * HIDP + kofs;
    const float* wb1 = wb0 + 16 * HIDP;
    v8f acc0 = {}, acc1 = {};
    for (int kb = 0; kb < HIDP; kb += 4) {
      v2f a = *(const v2f*)(HB + arow * HIDP + kb + kofs);
      acc0 = wmma4(a, *(const v2f*)(wb0 + kb), acc0);
      acc1 = wmma4(a, *(const v2f*)(wb1 + kb), acc1);
    }
    fc2_store(out, fc2_b, r0, n0,      arow, hi, acc0);
    fc2_store(out, fc2_b, r0, n0 + 16, arow, hi, acc1);
  }
}

extern "C" void kernel_launch(void* const* d_in, const int* in_sizes, int n_in,
                              void* d_out, int out_size, void* d_ws, size_t ws_size,
                              hipStream_t stream) {
  const float* cur = (const float*)d_in[0];
  const float* nxt = (const float*)d_in[1];
  const int*   act = (const int*)  d_in[2];
  const float* w1  = (const float*)d_in[3];
  const float* b1  = (const float*)d_in[4];
  const float* w2  = (const float*)d_in[5];
  const float* b2  = (const float*)d_in[6];
  const float* hw0 = (const float*)d_in[7];
  const float* hb0 = (const float*)d_in[8];
  const float* hw1 = (const float*)d_in[9];
  const float* hb1 = (const float*)d_in[10];
  const float* hw2 = (const float*)d_in[11];
  const float* hb2 = (const float*)d_in[12];
  float* out = (float*)d_out;

  const int n_tiles = NROWS / 16;        // 8192 row tiles
  const int grid    = n_tiles / BLKS;    // 2048 workgroups (8 waves each)
  const size_t smem = (size_t)LDS_FLOATS * sizeof(float);  // ~302.6 KB (<320 KB/WGP)

  icm_fused_kernel<<<grid, TPB, smem, stream>>>(
      cur, nxt, act, w1, b1, w2, b2, hw0, hb0, hw1, hb1, hw2, hb2, out);
}